// GCT_44968307589463
// MI455X (gfx1250) — compile-verified
//
#include <hip/hip_runtime.h>
#include <hip/hip_bf16.h>

#define EPSV 1e-5f

typedef float v2f __attribute__((ext_vector_type(2)));
typedef float v8f __attribute__((ext_vector_type(8)));

__device__ __forceinline__ v8f wmma4(v8f c, v2f a, v2f b) {
  // D = A(16x4 f32) * B(4x16 f32) + C(16x16 f32)
  return __builtin_amdgcn_wmma_f32_16x16x4_f32(false, a, false, b, (short)0, c, false, false);
}
__device__ __forceinline__ float lk(float x) { return x >= 0.f ? x : 0.2f * x; }
__device__ __forceinline__ unsigned fkey(float x) {
  int ib = __float_as_int(x);
  return (ib >= 0) ? (unsigned(ib) | 0x80000000u) : ~unsigned(ib);
}

constexpr int NB = 8, NN = 4096, KK = 20;
constexpr int PTSN = NB * NN; // 32768

// workspace offsets (in floats)
constexpr size_t OFF_FEATT = 0;
constexpr size_t OFF_PTST  = OFF_FEATT + (size_t)PTSN * 32;
constexpr size_t OFF_QPRE  = OFF_PTST + (size_t)PTSN * 4;
constexpr size_t OFF_P1PRE = OFF_QPRE + (size_t)PTSN * 64;
constexpr size_t OFF_Y     = OFF_P1PRE + (size_t)PTSN * 8;
constexpr size_t OFF_AF    = OFF_Y + (size_t)PTSN * KK * 16;
constexpr size_t OFF_BF    = OFF_AF + 2048;
constexpr size_t OFF_AP    = OFF_BF + 2048;
constexpr size_t OFF_BP    = OFF_AP + 32;
constexpr size_t OFF_QS    = OFF_BP + 32;
constexpr size_t OFF_QT    = OFF_QS + 64;
constexpr size_t OFF_PS    = OFF_QT + 64;
constexpr size_t OFF_PT    = OFF_PS + 8;
constexpr size_t OFF_YSC   = OFF_PT + 8;
constexpr size_t OFF_YSH   = OFF_YSC + 16;
constexpr size_t OFF_RSC   = OFF_YSH + 16;
constexpr size_t OFF_RSH   = OFF_RSC + 64;
constexpr size_t OFF_WLINP = OFF_RSH + 64;
constexpr size_t OFF_WRESP = OFF_WLINP + 1024;
constexpr size_t OFF_TSH   = OFF_WRESP + 4096;
constexpr size_t OFF_PART1 = OFF_TSH + 64;                    // 128 blocks * 140
constexpr size_t OFF_PART3 = OFF_PART1 + (size_t)128 * 140;   // 2048 blocks * 160
constexpr size_t OFF_PART5 = OFF_PART3 + (size_t)2048 * 160;  // 2560 blocks * 128

// ---------------- prep: fold efr weights ----------------
__global__ void k_prep(const float* __restrict__ wefp, const float* __restrict__ weff,
                       float* __restrict__ W) {
  int t = blockIdx.x * 256 + threadIdx.x;
  if (t < 2048) {
    int o = t >> 5, c = t & 31;
    float base = weff[o * 65 + c];
    W[OFF_AF + t] = base + weff[o * 65 + 32] * (1.f / 32.f);
    W[OFF_BF + t] = weff[o * 65 + 33 + c] - base;
  }
  if (t < 18) {
    int o = t / 3, c = t % 3;
    float base = wefp[o * 7 + c];
    W[OFF_AP + o * 4 + c] = base + wefp[o * 7 + 3] * (1.f / 3.f);
    W[OFF_BP + o * 4 + c] = wefp[o * 7 + 4 + c] - base;
  }
}

// ---------------- stage1: transposes, q_pre, pos1_pre + stats ----------------
__global__ __launch_bounds__(256) void k_stage1(const float* __restrict__ points,
                                                const float* __restrict__ feat,
                                                const float* __restrict__ w_q,
                                                const float* __restrict__ w_pos1,
                                                float* __restrict__ W) {
  int p = blockIdx.x * 256 + threadIdx.x; // 0..32767
  int b = p >> 12, n = p & 4095;
  float f[32];
#pragma unroll
  for (int c = 0; c < 32; ++c) {
    f[c] = feat[(size_t)(b * 32 + c) * NN + n];
    W[OFF_FEATT + (size_t)p * 32 + c] = f[c];
  }
  float pt3[3];
#pragma unroll
  for (int c = 0; c < 3; ++c) {
    pt3[c] = points[(size_t)(b * 3 + c) * NN + n];
    W[OFF_PTST + (size_t)p * 4 + c] = pt3[c];
  }
  __shared__ float red[140];
  if (threadIdx.x < 140) red[threadIdx.x] = 0.f;
  __syncthreads();
  int lane = threadIdx.x & 31;
  for (int o = 0; o < 64; ++o) {
    float q = 0.f;
#pragma unroll
    for (int c = 0; c < 32; ++c) q += w_q[o * 32 + c] * f[c];
    W[OFF_QPRE + (size_t)p * 64 + o] = q;
    float s = q, s2 = q * q;
    for (int m = 16; m; m >>= 1) { s += __shfl_xor(s, m, 32); s2 += __shfl_xor(s2, m, 32); }
    if (lane == 0) { atomicAdd(&red[o], s); atomicAdd(&red[64 + o], s2); }
  }
  for (int o = 0; o < 6; ++o) {
    float q = 0.f;
#pragma unroll
    for (int c = 0; c < 3; ++c) q += w_pos1[o * 3 + c] * pt3[c];
    W[OFF_P1PRE + (size_t)p * 8 + o] = q;
    float s = q, s2 = q * q;
    for (int m = 16; m; m >>= 1) { s += __shfl_xor(s, m, 32); s2 += __shfl_xor(s2, m, 32); }
    if (lane == 0) { atomicAdd(&red[128 + o], s); atomicAdd(&red[134 + o], s2); }
  }
  __syncthreads();
  if (threadIdx.x < 140) W[OFF_PART1 + (size_t)blockIdx.x * 140 + threadIdx.x] = red[threadIdx.x];
}

__global__ void k_red1(const float* __restrict__ g_q, const float* __restrict__ b_q,
                       const float* __restrict__ g_p, const float* __restrict__ b_p,
                       float* __restrict__ W) {
  int t = threadIdx.x;
  __shared__ float acc[140];
  if (t < 140) {
    float s = 0.f;
    for (int blk = 0; blk < 128; ++blk) s += W[OFF_PART1 + (size_t)blk * 140 + t];
    acc[t] = s;
  }
  __syncthreads();
  const float invM = 1.f / 32768.f;
  if (t < 64) {
    float m = acc[t] * invM, v = acc[64 + t] * invM - m * m;
    float sc = g_q[t] * rsqrtf(v + EPSV);
    W[OFF_QS + t] = sc; W[OFF_QT + t] = b_q[t] - m * sc;
  } else if (t < 70) {
    int c = t - 64;
    float m = acc[128 + c] * invM, v = acc[134 + c] * invM - m * m;
    float sc = g_p[c] * rsqrtf(v + EPSV);
    W[OFF_PS + c] = sc; W[OFF_PT + c] = b_p[c] - m * sc;
  }
}

// ---------------- stage3: fused attention tile kernel ----------------
__global__ __launch_bounds__(256) void k_stage3(const int* __restrict__ idx,
                                                const float* __restrict__ w_res,
                                                const float* __restrict__ w_attn,
                                                const float* __restrict__ w_pos2,
                                                float* __restrict__ W) {
  __shared__ float sQ[16 * 64];
  __shared__ float sPosL[16 * 8];
  __shared__ float sCtrF[16 * 32];
  __shared__ float sCtrP[16 * 4];
  __shared__ float sNbrP[320 * 4];
  __shared__ float sKnnP[320 * 8];
  __shared__ float sCtrKF[16 * 64];
  __shared__ float sAf[64 * 32];
  __shared__ float sW[96 * 64];
  __shared__ float sWp2[64 * 8];
  __shared__ float sKF[320 * 64];
  __shared__ float sA96[80 * 96];
  __shared__ float sStat[160];
  __shared__ int sIdx[320];

  const int tid = threadIdx.x;
  const int wv = tid >> 5, lane = tid & 31, lnN = lane & 15, hf = lane >> 4;
  const int blk = blockIdx.x;
  const int b = blk >> 8;
  const int n0 = (blk & 255) << 4;
  const int p0 = b * NN + n0;
  const float* FT = W + OFF_FEATT;
  const float* PTt = W + OFF_PTST;

  // Phase A: loads
  for (int i = tid; i < 320; i += 256) sIdx[i] = idx[(size_t)(p0 + i / 20) * KK + (i % 20)];
  for (int i = tid; i < 1024; i += 256) {
    int nl = i >> 6, o = i & 63;
    sQ[i] = lk(W[OFF_QS + o] * W[OFF_QPRE + (size_t)(p0 + nl) * 64 + o] + W[OFF_QT + o]);
  }
  for (int i = tid; i < 96; i += 256) {
    int nl = i / 6, c = i % 6;
    sPosL[nl * 8 + c] = lk(W[OFF_PS + c] * W[OFF_P1PRE + (size_t)(p0 + nl) * 8 + c] + W[OFF_PT + c]);
  }
  for (int i = tid; i < 512; i += 256) { int nl = i >> 5, c = i & 31; sCtrF[i] = FT[(size_t)(p0 + nl) * 32 + c]; }
  for (int i = tid; i < 48; i += 256) { int nl = i / 3, c = i % 3; sCtrP[nl * 4 + c] = PTt[(size_t)(p0 + nl) * 4 + c]; }
  for (int i = tid; i < 2048; i += 256) sAf[i] = W[OFF_AF + i];
  for (int i = tid; i < 4096; i += 256) sW[i] = w_res[i];
  for (int i = tid; i < 384; i += 256) { int o = i / 6, c = i % 6; sWp2[o * 8 + c] = w_pos2[i]; }
  if (tid < 160) sStat[tid] = 0.f;
  __syncthreads();

  // Phase B: gather neighbor points; Bf @ ctr_feat
  for (int i = tid; i < 960; i += 256) { int s = i / 3, c = i % 3; sNbrP[s * 4 + c] = PTt[(size_t)sIdx[s] * 4 + c]; }
  for (int i = tid; i < 1024; i += 256) {
    int nl = i >> 6, o = i & 63;
    const float* bf = &W[OFF_BF + o * 32];
    const float* cf = &sCtrF[nl * 32];
    float a = 0.f;
#pragma unroll
    for (int c = 0; c < 32; ++c) a += bf[c] * cf[c];
    sCtrKF[i] = a;
  }
  __syncthreads();

  // Phase C: knn_points (VALU) + knn_feat (WMMA, A gathered from global featT)
  for (int i = tid; i < 1920; i += 256) {
    int s = i / 6, o = i % 6, nl = s / 20;
    float a = 0.f;
#pragma unroll
    for (int c = 0; c < 3; ++c)
      a += W[OFF_AP + o * 4 + c] * sNbrP[s * 4 + c] + W[OFF_BP + o * 4 + c] * sCtrP[nl * 4 + c];
    sKnnP[s * 8 + o] = a;
  }
  for (int t = wv; t < 80; t += 8) {
    int st = t >> 2, ft = t & 3;
    const float* ap = FT + (size_t)sIdx[st * 16 + lnN] * 32 + 2 * hf;
    const float* bp = &sAf[(ft * 16 + lnN) * 32 + 2 * hf];
    v8f c = {};
#pragma unroll
    for (int kc = 0; kc < 8; ++kc)
      c = wmma4(c, *(const v2f*)(ap + 4 * kc), *(const v2f*)(bp + 4 * kc));
    int f = ft * 16 + lnN;
#pragma unroll
    for (int v = 0; v < 8; ++v) {
      int s = st * 16 + v + hf * 8;
      sKF[s * 64 + f] = c[v] + sCtrKF[(s / 20) * 64 + f];
    }
  }
  __syncthreads();

  // Phase D: res_pre = W_res @ knn_feat (WMMA) -> channel stats only
  for (int t = wv; t < 80; t += 8) {
    int st = t >> 2, ot = t & 3;
    const float* ap = &sKF[(st * 16 + lnN) * 64 + 2 * hf];
    const float* bp = &sW[(ot * 16 + lnN) * 64 + 2 * hf];
    v8f c = {};
#pragma unroll
    for (int kc = 0; kc < 16; ++kc)
      c = wmma4(c, *(const v2f*)(ap + 4 * kc), *(const v2f*)(bp + 4 * kc));
    float s1 = 0.f, s2 = 0.f;
#pragma unroll
    for (int v = 0; v < 8; ++v) { float x = c[v]; s1 += x; s2 += x * x; }
    int o = ot * 16 + lnN;
    atomicAdd(&sStat[o], s1);
    atomicAdd(&sStat[64 + o], s2);
  }
  __syncthreads();

  // Phase E: overwrite sKF with attn logits; swap sW -> w_attn
  for (int i = tid; i < 6144; i += 256) sW[i] = w_attn[i];
  for (int i = tid; i < 320 * 64; i += 256) {
    int s = i >> 6, ch = i & 63, nl = s / 20;
    float pos = 0.f;
#pragma unroll
    for (int c = 0; c < 6; ++c) pos += sWp2[ch * 8 + c] * (sPosL[nl * 8 + c] - sKnnP[s * 8 + c]);
    sKF[i] = sQ[nl * 64 + ch] - sKF[i] + pos;
  }
  __syncthreads();

  // Phase F: softmax over K (per (n, channel))
  for (int i = tid; i < 1024; i += 256) {
    int nl = i >> 6, ch = i & 63;
    float* col = &sKF[(nl * 20) * 64 + ch];
    float mx = -3.4e38f;
#pragma unroll
    for (int k = 0; k < 20; ++k) mx = fmaxf(mx, col[k * 64] * 0.125f);
    float e[20], sum = 0.f;
#pragma unroll
    for (int k = 0; k < 20; ++k) { e[k] = __expf(col[k * 64] * 0.125f - mx); sum += e[k]; }
    float inv = 1.f / sum;
#pragma unroll
    for (int k = 0; k < 20; ++k) col[k * 64] = e[k] * inv;
  }
  __syncthreads();

  // Phase G: attn96 = W_attn @ attn (WMMA, chunked) -> y = sum_c attn96[6h+c]*knnP[c]
  float* Wy = W + OFF_Y;
  float ysl = 0.f, yql = 0.f;
  for (int cc = 0; cc < 4; ++cc) {
    for (int t = wv; t < 30; t += 8) {
      int stc = t / 6, at = t % 6;
      const float* ap = &sKF[(cc * 80 + stc * 16 + lnN) * 64 + 2 * hf];
      const float* bp = &sW[(at * 16 + lnN) * 64 + 2 * hf];
      v8f c = {};
#pragma unroll
      for (int kc = 0; kc < 16; ++kc)
        c = wmma4(c, *(const v2f*)(ap + 4 * kc), *(const v2f*)(bp + 4 * kc));
      int a = at * 16 + lnN;
#pragma unroll
      for (int v = 0; v < 8; ++v) sA96[(stc * 16 + v + hf * 8) * 96 + a] = c[v];
    }
    __syncthreads();
    for (int i = tid; i < 1280; i += 256) {
      int sl = i >> 4, h = i & 15;
      int s = cc * 80 + sl, nl = s / 20, kq = s % 20;
      float y = 0.f;
#pragma unroll
      for (int c = 0; c < 6; ++c) y += sA96[sl * 96 + h * 6 + c] * sKnnP[s * 8 + c];
      Wy[((size_t)(p0 + nl) * KK + kq) * 16 + h] = y;
      ysl += y; yql += y * y;
    }
    __syncthreads();
  }
  ysl += __shfl_xor(ysl, 16, 32);
  yql += __shfl_xor(yql, 16, 32);
  if (lane < 16) { atomicAdd(&sStat[128 + (tid & 15)], ysl); atomicAdd(&sStat[144 + (tid & 15)], yql); }
  __syncthreads();
  if (tid < 160) W[OFF_PART3 + (size_t)blk * 160 + tid] = sStat[tid];
}

__global__ void k_red3(const float* __restrict__ g1, const float* __restrict__ b1,
                       const float* __restrict__ g_res, const float* __restrict__ b_res,
                       float* __restrict__ W) {
  int t = threadIdx.x;
  __shared__ float acc[160];
  if (t < 160) {
    float s = 0.f;
    for (int k = 0; k < 2048; ++k) s += W[OFF_PART3 + (size_t)k * 160 + t];
    acc[t] = s;
  }
  __syncthreads();
  const float invM = 1.f / 655360.f;
  if (t < 64) {
    float m = acc[t] * invM, v = acc[64 + t] * invM - m * m;
    float sc = g_res[t] * rsqrtf(v + EPSV);
    W[OFF_RSC + t] = sc; W[OFF_RSH + t] = b_res[t] - m * sc;
  } else if (t < 80) {
    int h = t - 64;
    float m = acc[128 + h] * invM, v = acc[144 + h] * invM - m * m;
    float sc = g1[h] * rsqrtf(v + EPSV);
    W[OFF_YSC + h] = sc; W[OFF_YSH + h] = b1[h] - m * sc;
  }
}

// ---------------- stage5: stats of W_lin @ leaky(bn(y)) ----------------
__global__ __launch_bounds__(256) void k_stage5(const float* __restrict__ w_lin,
                                                float* __restrict__ W) {
  __shared__ float sWL[1024];
  __shared__ float red[128];
  __shared__ float ysc[16], ysh[16];
  int tid = threadIdx.x;
  for (int i = tid; i < 1024; i += 256) sWL[i] = w_lin[i];
  if (tid < 128) red[tid] = 0.f;
  if (tid < 16) { ysc[tid] = W[OFF_YSC + tid]; ysh[tid] = W[OFF_YSH + tid]; }
  __syncthreads();
  size_t gs = (size_t)blockIdx.x * 256 + tid;
  const float* yp = W + OFF_Y + gs * 16;
  float z[16];
#pragma unroll
  for (int h = 0; h < 16; ++h) z[h] = lk(ysc[h] * yp[h] + ysh[h]);
  int lane = tid & 31;
  for (int o = 0; o < 64; ++o) {
    float v = 0.f;
#pragma unroll
    for (int h = 0; h < 16; ++h) v += sWL[o * 16 + h] * z[h];
    float s = v, s2 = v * v;
    for (int m = 16; m; m >>= 1) { s += __shfl_xor(s, m, 32); s2 += __shfl_xor(s2, m, 32); }
    if (lane == 0) { atomicAdd(&red[o], s); atomicAdd(&red[64 + o], s2); }
  }
  __syncthreads();
  if (tid < 128) W[OFF_PART5 + (size_t)blockIdx.x * 128 + tid] = red[tid];
}

__global__ void k_red5(const float* __restrict__ g_lin, const float* __restrict__ b_lin,
                       const float* __restrict__ w_lin, const float* __restrict__ w_res,
                       float* __restrict__ W) {
  int t = threadIdx.x;
  __shared__ float acc[128];
  __shared__ float lsc[64], lsh[64];
  if (t < 128) {
    float s = 0.f;
    for (int k = 0; k < 2560; ++k) s += W[OFF_PART5 + (size_t)k * 128 + t];
    acc[t] = s;
  }
  __syncthreads();
  const float invM = 1.f / 655360.f;
  if (t < 64) {
    float m = acc[t] * invM, v = acc[64 + t] * invM - m * m;
    float sc = g_lin[t] * rsqrtf(v + EPSV);
    lsc[t] = sc; lsh[t] = b_lin[t] - m * sc;
  }
  __syncthreads();
  if (t < 64) {
    float sc = lsc[t];
    for (int h = 0; h < 16; ++h) W[OFF_WLINP + t * 16 + h] = sc * w_lin[t * 16 + h];
    float rc = W[OFF_RSC + t];
    for (int c = 0; c < 64; ++c) W[OFF_WRESP + t * 64 + c] = rc * w_res[t * 64 + c];
    W[OFF_TSH + t] = lsh[t] + W[OFF_RSH + t];
  }
}

// ---------------- stage7: final fused output (combined WMMA + K-max) ----------------
__global__ __launch_bounds__(256) void k_stage7(const int* __restrict__ idx,
                                                float* __restrict__ out,
                                                float* __restrict__ W) {
  __shared__ float sCtrF[16 * 32];
  __shared__ float sCtrKF[16 * 64];
  __shared__ float sAf[64 * 32];
  __shared__ float sWR[64 * 64];
  __shared__ float sWL[64 * 16];
  __shared__ float sTsh[64];
  __shared__ float sKF[320 * 64];
  __shared__ float sZ[320 * 16];
  __shared__ unsigned sMax[16 * 64];
  __shared__ int sIdx[320];

  const int tid = threadIdx.x;
  const int wv = tid >> 5, lane = tid & 31, lnN = lane & 15, hf = lane >> 4;
  const int blk = blockIdx.x;
  const int b = blk >> 8;
  const int n0 = (blk & 255) << 4;
  const int p0 = b * NN + n0;
  const float* FT = W + OFF_FEATT;

  for (int i = tid; i < 320; i += 256) sIdx[i] = idx[(size_t)(p0 + i / 20) * KK + (i % 20)];
  for (int i = tid; i < 512; i += 256) { int nl = i >> 5, c = i & 31; sCtrF[i] = FT[(size_t)(p0 + nl) * 32 + c]; }
  for (int i = tid; i < 2048; i += 256) sAf[i] = W[OFF_AF + i];
  for (int i = tid; i < 4096; i += 256) sWR[i] = W[OFF_WRESP + i];
  for (int i = tid; i < 1024; i += 256) sWL[i] = W[OFF_WLINP + i];
  if (tid < 64) sTsh[tid] = W[OFF_TSH + tid];
  for (int i = tid; i < 1024; i += 256) sMax[i] = 0u;
  for (int i = tid; i < 5120; i += 256) {
    int s = i >> 4, h = i & 15, nl = s / 20, kq = s % 20;
    float y = W[OFF_Y + ((size_t)(p0 + nl) * KK + kq) * 16 + h];
    sZ[i] = lk(W[OFF_YSC + h] * y + W[OFF_YSH + h]);
  }
  __syncthreads();
  for (int i = tid; i < 1024; i += 256) {
    int nl = i >> 6, o = i & 63;
    const float* bf = &W[OFF_BF + o * 32];
    const float* cf = &sCtrF[nl * 32];
    float a = 0.f;
#pragma unroll
    for (int c = 0; c < 32; ++c) a += bf[c] * cf[c];
    sCtrKF[i] = a;
  }
  __syncthreads();
  for (int t = wv; t < 80; t += 8) {
    int st = t >> 2, ft = t & 3;
    const float* ap = FT + (size_t)sIdx[st * 16 + lnN] * 32 + 2 * hf;
    const float* bp = &sAf[(ft * 16 + lnN) * 32 + 2 * hf];
    v8f c = {};
#pragma unroll
    for (int kc = 0; kc < 8; ++kc)
      c = wmma4(c, *(const v2f*)(ap + 4 * kc), *(const v2f*)(bp + 4 * kc));
    int f = ft * 16 + lnN;
#pragma unroll
    for (int v = 0; v < 8; ++v) {
      int s = st * 16 + v + hf * 8;
      sKF[s * 64 + f] = c[v] + sCtrKF[(s / 20) * 64 + f];
    }
  }
  __syncthreads();
  for (int t = wv; t < 80; t += 8) {
    int st = t >> 2, ot = t & 3;
    int srow = st * 16 + lnN;
    v8f c = {};
    const float* ap = &sKF[srow * 64 + 2 * hf];
    const float* bp = &sWR[(ot * 16 + lnN) * 64 + 2 * hf];
#pragma unroll
    for (int kc = 0; kc < 16; ++kc)
      c = wmma4(c, *(const v2f*)(ap + 4 * kc), *(const v2f*)(bp + 4 * kc));
    const float* az = &sZ[srow * 16 + 2 * hf];
    const float* bz = &sWL[(ot * 16 + lnN) * 16 + 2 * hf];
#pragma unroll
    for (int kc = 0; kc < 4; ++kc)
      c = wmma4(c, *(const v2f*)(az + 4 * kc), *(const v2f*)(bz + 4 * kc));
    int o = ot * 16 + lnN;
    float tsh = sTsh[o];
#pragma unroll
    for (int v = 0; v < 8; ++v) {
      int s = st * 16 + v + hf * 8;
      float x = lk(c[v] + tsh);
      atomicMax(&sMax[(s / 20) * 64 + o], fkey(x));
    }
  }
  __syncthreads();
  for (int i = tid; i < 1024; i += 256) {
    int nl = i >> 6, o = i & 63;
    unsigned u = sMax[i];
    float v = (u & 0x80000000u) ? __int_as_float(int(u & 0x7fffffffu)) : __int_as_float(int(~u));
    out[((size_t)b * 64 + o) * NN + n0 + nl] = v;
  }
}

extern "C" void kernel_launch(void* const* d_in, const int* in_sizes, int n_in,
                              void* d_out, int out_size, void* d_ws, size_t ws_size,
                              hipStream_t stream) {
  const float* points = (const float*)d_in[0];
  const float* feat   = (const float*)d_in[1];
  const int*   idx    = (const int*)d_in[2];
  const float* wefp   = (const float*)d_in[3];
  const float* weff   = (const float*)d_in[4];
  const float* w_q    = (const float*)d_in[5];
  const float* g_q    = (const float*)d_in[6];
  const float* b_q    = (const float*)d_in[7];
  const float* w_pos1 = (const float*)d_in[8];
  const float* g_pos1 = (const float*)d_in[9];
  const float* b_pos1 = (const float*)d_in[10];
  const float* w_pos2 = (const float*)d_in[11];
  const float* w_attn = (const float*)d_in[12];
  const float* g1     = (const float*)d_in[13];
  const float* b1     = (const float*)d_in[14];
  const float* w_lin  = (const float*)d_in[15];
  const float* g_lin  = (const float*)d_in[16];
  const float* b_lin  = (const float*)d_in[17];
  const float* w_res  = (const float*)d_in[18];
  const float* g_res  = (const float*)d_in[19];
  const float* b_res  = (const float*)d_in[20];
  float* W = (float*)d_ws;
  float* out = (float*)d_out;

  k_prep<<<8, 256, 0, stream>>>(wefp, weff, W);
  k_stage1<<<128, 256, 0, stream>>>(points, feat, w_q, w_pos1, W);
  k_red1<<<1, 160, 0, stream>>>(g_q, b_q, g_pos1, b_pos1, W);
  k_stage3<<<2048, 256, 0, stream>>>(idx, w_res, w_attn, w_pos2, W);
  k_red3<<<1, 160, 0, stream>>>(g1, b1, g_res, b_res, W);
  k_stage5<<<2560, 256, 0, stream>>>(w_lin, W);
  k_red5<<<1, 128, 0, stream>>>(g_lin, b_lin, w_lin, w_res, W);
  k_stage7<<<2048, 256, 0, stream>>>(idx, out, W);
  (void)in_sizes; (void)n_in; (void)out_size; (void)ws_size;
}